// GCN_55972013802295
// MI455X (gfx1250) — compile-verified
//
#include <hip/hip_runtime.h>
#include <math.h>

typedef __attribute__((ext_vector_type(2))) float v2f;
typedef __attribute__((ext_vector_type(8))) float v8f;

#define HIDF 128

// ---------------- degree / normalization ----------------
__global__ void k_init_deg(float* __restrict__ deg, int n) {
  int i = blockIdx.x * blockDim.x + threadIdx.x;
  if (i < n) deg[i] = 1.0f;  // self-loop weight 1
}

__global__ void k_edge_deg(const int* __restrict__ col, const float* __restrict__ ew,
                           float* __restrict__ deg, int E) {
  int e = blockIdx.x * blockDim.x + threadIdx.x;
  if (e < E) atomicAdd(&deg[col[e]], ew[e]);
}

__global__ void k_rsqrt(float* __restrict__ deg, int n) {
  int i = blockIdx.x * blockDim.x + threadIdx.x;
  if (i < n) {
    float d = deg[i];
    deg[i] = (d > 0.0f) ? rsqrtf(d) : 0.0f;
  }
}

// ---------------- fp32 WMMA GEMM: C[M x 128] = A[M x 128] @ W[128 x 128] (+bias) ----------------
// One block = 256 threads = 8 waves. Block owns a 16-row M tile; wave w owns
// N-columns [16w, 16w+16). K loop: 32 steps of V_WMMA_F32_16X16X4_F32.
// M (=100000) is a multiple of 16, so no partial tiles -> EXEC all-1s at WMMA.
__global__ __launch_bounds__(256)
void k_gemm_wmma(const float* __restrict__ A, const float* __restrict__ W,
                 float* __restrict__ C, const float* __restrict__ bias) {
  // A tile in LDS, padded pitch 132 floats (132 % 64 = 4 banks offset per row)
  __shared__ float As[16 * 132];
  const int t  = threadIdx.x;
  const int m0 = blockIdx.x * 16;

  // stage 16x128 A tile (512 float4s, 2 per thread)
  {
    const float4* A4 = (const float4*)(A + (size_t)m0 * HIDF);
    for (int i = t; i < 16 * 32; i += 256) {
      int r = i >> 5, c = i & 31;
      float4 v = A4[i];
      As[r * 132 + c * 4 + 0] = v.x;
      As[r * 132 + c * 4 + 1] = v.y;
      As[r * 132 + c * 4 + 2] = v.z;
      As[r * 132 + c * 4 + 3] = v.w;
    }
  }
  __syncthreads();

  const int lane = t & 31;
  const int wave = t >> 5;
  const int half = lane >> 4;   // 0: K pair {k0,k0+1}; 1: {k0+2,k0+3}
  const int l15  = lane & 15;
  const int n0   = wave * 16;

  v8f acc = {};
#pragma unroll
  for (int kt = 0; kt < 32; ++kt) {
    const int k0 = kt * 4 + 2 * half;
#if defined(__has_builtin) && __has_builtin(__builtin_amdgcn_wmma_f32_16x16x4_f32)
    v2f a, b;
    // A 16x4 frag (ISA layout): lane<16 -> M=lane, K=k0..k0+1; lane>=16 -> K=k0+2..k0+3
    a.x = As[l15 * 132 + k0 + 0];
    a.y = As[l15 * 132 + k0 + 1];
    // B 4x16 frag: N = n0+l15, K rows mirror A's half split
    b.x = W[(k0 + 0) * HIDF + n0 + l15];
    b.y = W[(k0 + 1) * HIDF + n0 + l15];
    acc = __builtin_amdgcn_wmma_f32_16x16x4_f32(false, a, false, b, (short)0, acc,
                                                false, false);
#else
    // scalar fallback with identical output mapping
#pragma unroll
    for (int r = 0; r < 8; ++r) {
      int M = r + 8 * half;
      int kk = kt * 4;
      float s = acc[r];
      s += As[M * 132 + kk + 0] * W[(kk + 0) * HIDF + n0 + l15];
      s += As[M * 132 + kk + 1] * W[(kk + 1) * HIDF + n0 + l15];
      s += As[M * 132 + kk + 2] * W[(kk + 2) * HIDF + n0 + l15];
      s += As[M * 132 + kk + 3] * W[(kk + 3) * HIDF + n0 + l15];
      acc[r] = s;
    }
#endif
  }

  const float bv = bias ? bias[n0 + l15] : 0.0f;
  // D layout: VGPR r -> M = r + 8*half, N = l15
#pragma unroll
  for (int r = 0; r < 8; ++r) {
    C[(size_t)(m0 + r + 8 * half) * HIDF + (n0 + l15)] = acc[r] + bv;
  }
}

// ---------------- self-loop init: agg = hw * dis^2 ----------------
__global__ void k_selfagg(const float* __restrict__ hw, const float* __restrict__ dis,
                          float* __restrict__ agg, int n) {
  int idx = blockIdx.x * blockDim.x + threadIdx.x;  // one float4 each, n*32 total
  if (idx >= n * 32) return;
  int node = idx >> 5;
  float s = dis[node];
  s = s * s;
  float4 v = ((const float4*)hw)[idx];
  float4 o;
  o.x = v.x * s; o.y = v.y * s; o.z = v.z * s; o.w = v.w * s;
  ((float4*)agg)[idx] = o;
}

// ---------------- edge scatter: agg[col] += norm * hw[row] ----------------
// One wave (32 lanes) per edge; each lane moves one float4 (4 fp32 atomics).
__global__ __launch_bounds__(256)
void k_scatter(const float* __restrict__ hw, const int* __restrict__ row,
               const int* __restrict__ col, const float* __restrict__ ew,
               const float* __restrict__ dis, float* __restrict__ agg, int E) {
  int e = blockIdx.x * 8 + (threadIdx.x >> 5);
  if (e >= E) return;
  const int lane = threadIdx.x & 31;
  const int r = row[e];
  const int c = col[e];
  const float norm = dis[r] * ew[e] * dis[c];
  float4 v = ((const float4*)(hw + (size_t)r * HIDF))[lane];
  float* dst = agg + (size_t)c * HIDF + lane * 4;
  atomicAdd(dst + 0, v.x * norm);
  atomicAdd(dst + 1, v.y * norm);
  atomicAdd(dst + 2, v.z * norm);
  atomicAdd(dst + 3, v.w * norm);
}

// ---------------- bias + optional ReLU ----------------
__global__ void k_finalize(const float* __restrict__ agg, const float* __restrict__ b,
                           float* __restrict__ out, int n, int relu) {
  int idx = blockIdx.x * blockDim.x + threadIdx.x;  // one float4 each
  if (idx >= n * 32) return;
  int c4 = idx & 31;
  float4 v  = ((const float4*)agg)[idx];
  float4 bb = ((const float4*)b)[c4];
  v.x += bb.x; v.y += bb.y; v.z += bb.z; v.w += bb.w;
  if (relu) {
    v.x = fmaxf(v.x, 0.0f); v.y = fmaxf(v.y, 0.0f);
    v.z = fmaxf(v.z, 0.0f); v.w = fmaxf(v.w, 0.0f);
  }
  ((float4*)out)[idx] = v;
}

extern "C" void kernel_launch(void* const* d_in, const int* in_sizes, int n_in,
                              void* d_out, int out_size, void* d_ws, size_t ws_size,
                              hipStream_t stream) {
  const float* x   = (const float*)d_in[0];
  const int*   ei  = (const int*)d_in[1];
  const float* ew  = (const float*)d_in[2];
  const float* Wc[4] = {(const float*)d_in[3], (const float*)d_in[5],
                        (const float*)d_in[7], (const float*)d_in[9]};
  const float* bc[4] = {(const float*)d_in[4], (const float*)d_in[6],
                        (const float*)d_in[8], (const float*)d_in[10]};
  const float* Wout = (const float*)d_in[11];
  const float* bout = (const float*)d_in[12];

  const int n = in_sizes[0] / HIDF;   // 100000
  const int E = in_sizes[1] / 2;      // 1600000
  const int* row = ei;
  const int* col = ei + E;

  // workspace layout: [dis (n floats)] [hw] [agg] [hbuf]  (each n*128 floats)
  char* ws = (char*)d_ws;
  float* dis = (float*)ws;
  size_t off = (((size_t)n * 4) + 255) & ~(size_t)255;
  float* hw   = (float*)(ws + off);
  float* agg  = hw  + (size_t)n * HIDF;
  float* hbuf = agg + (size_t)n * HIDF;

  // degree + d^{-1/2}
  k_init_deg<<<(n + 255) / 256, 256, 0, stream>>>(dis, n);
  k_edge_deg<<<(E + 255) / 256, 256, 0, stream>>>(col, ew, dis, E);
  k_rsqrt<<<(n + 255) / 256, 256, 0, stream>>>(dis, n);

  const int nf4 = n * 32;  // float4 count per feature map
  const float* hcur = x;
  for (int l = 0; l < 4; ++l) {
    k_gemm_wmma<<<n / 16, 256, 0, stream>>>(hcur, Wc[l], hw, nullptr);
    k_selfagg<<<(nf4 + 255) / 256, 256, 0, stream>>>(hw, dis, agg, n);
    k_scatter<<<(E + 7) / 8, 256, 0, stream>>>(hw, row, col, ew, dis, agg, E);
    k_finalize<<<(nf4 + 255) / 256, 256, 0, stream>>>(agg, bc[l], hbuf, n,
                                                      (l < 3) ? 1 : 0);
    hcur = hbuf;
  }
  // output projection with fused bias, straight into d_out
  k_gemm_wmma<<<n / 16, 256, 0, stream>>>(hcur, Wout, (float*)d_out, bout);
}